// CustomAnyDense_36283883717256
// MI455X (gfx1250) — compile-verified
//
#include <hip/hip_runtime.h>
#include <stdint.h>

// Problem sizes (fixed by the reference)
#define M_DIM 1024
#define K_DIM 2048
#define N_DIM 2048

// Tiling
#define BM 64
#define BN 128
#define KB 32
#define NTHREADS 256

// LDS layout (byte offsets). LUT must start at 0 so gather addr == (a<<8)|b.
#define LUT_BYTES  65536
#define AF_BYTES   (BM * KB * 4)          // 8192 per buffer
#define BF_BYTES   (BN * KB * 4)          // 16384 per buffer
#define AF_OFF     LUT_BYTES
#define BF_OFF     (AF_OFF + 2 * AF_BYTES)
#define AS_OFF     (BF_OFF + 2 * BF_BYTES)
#define BS_OFF     (AS_OFF + KB * BM)     // As: 2048 B
#define SMEM_BYTES (BS_OFF + KB * BN)     // Bs: 4096 B -> total 120832 B

__device__ __forceinline__ void async_b128(uint32_t lds_addr, uint32_t goff, uint64_t sbase) {
    // CDNA5 async DMA: global -> LDS, 16B per lane, tracked by ASYNCcnt.
    asm volatile("global_load_async_to_lds_b128 %0, %1, %2"
                 :: "v"(lds_addr), "v"(goff), "s"(sbase) : "memory");
}

__device__ __forceinline__ void wait_async0() {
    asm volatile("s_wait_asynccnt 0x0" ::: "memory");
}

__global__ __launch_bounds__(NTHREADS, 2)
void lut_dense_kernel(const float* __restrict__ A,      // [M, K] code floats
                      const float* __restrict__ Wt,     // [N, K] code floats (weight)
                      const float* __restrict__ bias,   // [N]
                      const int*   __restrict__ lut,    // [256*256] int32 (values 0..255)
                      float* __restrict__ out)          // [M, N]
{
    extern __shared__ unsigned char smem[];
    const int t  = threadIdx.x;
    const int n0 = blockIdx.x * BN;
    const int m0 = blockIdx.y * BM;

    const uint64_t abase = (uint64_t)(uintptr_t)A;
    const uint64_t wbase = (uint64_t)(uintptr_t)Wt;

    // ---- 1) LUT int32 -> packed u8 into the first 64KB of LDS ----
    {
        const int4* lut4 = (const int4*)lut;
        uint32_t*   lutw = (uint32_t*)(smem);   // 16384 words
        #pragma unroll 4
        for (int it = 0; it < 64; ++it) {
            int idx = it * NTHREADS + t;
            int4 v  = lut4[idx];
            uint32_t p = ((uint32_t)v.x & 255u)
                       | (((uint32_t)v.y & 255u) << 8)
                       | (((uint32_t)v.z & 255u) << 16)
                       | (((uint32_t)v.w & 255u) << 24);
            lutw[idx] = p;
        }
    }

    // ---- 2) async DMA of one K-chunk of A/B code floats into staging buffers ----
    auto kick = [&](int buf, int kc) {
        // A tile: BM rows x KB floats = 512 x 16B transfers
        #pragma unroll
        for (int r = 0; r < 2; ++r) {
            int e = t + r * NTHREADS;
            int row = e >> 3, seg = e & 7;
            uint32_t goff = (uint32_t)((((m0 + row) * K_DIM + kc) << 2) + (seg << 4));
            uint32_t lds  = (uint32_t)(uintptr_t)(smem + AF_OFF + buf * AF_BYTES
                                                  + row * (KB * 4) + seg * 16);
            async_b128(lds, goff, abase);
        }
        // B tile: BN rows x KB floats = 1024 x 16B transfers
        #pragma unroll
        for (int r = 0; r < 4; ++r) {
            int e = t + r * NTHREADS;
            int row = e >> 3, seg = e & 7;
            uint32_t goff = (uint32_t)((((n0 + row) * K_DIM + kc) << 2) + (seg << 4));
            uint32_t lds  = (uint32_t)(uintptr_t)(smem + BF_OFF + buf * BF_BYTES
                                                  + row * (KB * 4) + seg * 16);
            async_b128(lds, goff, wbase);
        }
    };

    kick(0, 0);

    const int tx = t & 15;          // 16 threads over N
    const int ty = t >> 4;          // 16 threads over M
    const int mb = ty * 4;          // row offset in tile
    const int nb = tx * 8;          // col offset in tile

    unsigned char* lutb = smem;             // 64KB-aligned at offset 0
    unsigned char* As   = smem + AS_OFF;    // [KB][BM] u8
    unsigned char* Bs   = smem + BS_OFF;    // [KB][BN] u8

    uint32_t acc[4][8];
    #pragma unroll
    for (int i = 0; i < 4; ++i)
        #pragma unroll
        for (int j = 0; j < 8; ++j) acc[i][j] = 0u;

    const int KCH = K_DIM / KB;     // 64 chunks
    for (int c = 0; c < KCH; ++c) {
        const int cur = c & 1;
        wait_async0();              // chunk c staged (this wave's DMAs)
        __syncthreads();            // all waves staged; prior compute done

        if (c + 1 < KCH) kick(cur ^ 1, (c + 1) * KB);   // overlap next DMA

        // ---- convert staged floats -> packed u8 code arrays (transposed) ----
        const float* Afc = (const float*)(smem + AF_OFF + cur * AF_BYTES);
        const float* Bfc = (const float*)(smem + BF_OFF + cur * BF_BYTES);
        #pragma unroll
        for (int r = 0; r < 2; ++r) {               // As: 512 words
            int w = t + r * NTHREADS;
            int k = w & (KB - 1);
            int mg = w >> 5;                        // 0..15
            uint32_t p = 0;
            #pragma unroll
            for (int i = 0; i < 4; ++i) {
                float f = Afc[(mg * 4 + i) * KB + k];
                uint32_t cd = (uint32_t)fminf(fmaxf(f, 0.0f), 255.0f);
                p |= cd << (8 * i);
            }
            ((uint32_t*)As)[k * (BM / 4) + mg] = p;
        }
        #pragma unroll
        for (int r = 0; r < 4; ++r) {               // Bs: 1024 words
            int w = t + r * NTHREADS;
            int k = w & (KB - 1);
            int ng = w >> 5;                        // 0..31
            uint32_t p = 0;
            #pragma unroll
            for (int i = 0; i < 4; ++i) {
                float f = Bfc[(ng * 4 + i) * KB + k];
                uint32_t cd = (uint32_t)fminf(fmaxf(f, 0.0f), 255.0f);
                p |= cd << (8 * i);
            }
            ((uint32_t*)Bs)[k * (BN / 4) + ng] = p;
        }
        __syncthreads();

        // ---- gather-accumulate: the hot loop ----
        #pragma unroll 2
        for (int k = 0; k < KB; ++k) {
            uint32_t a4   = *(const uint32_t*)(As + k * BM + mb);
            uint32_t b4lo = *(const uint32_t*)(Bs + k * BN + nb);
            uint32_t b4hi = *(const uint32_t*)(Bs + k * BN + nb + 4);
            #pragma unroll
            for (int i = 0; i < 4; ++i) {
                uint32_t ab = ((a4 >> (8 * i)) & 0xFFu) << 8;
                #pragma unroll
                for (int j = 0; j < 4; ++j) {
                    acc[i][j]     += (uint32_t)lutb[ab | ((b4lo >> (8 * j)) & 0xFFu)];
                    acc[i][j + 4] += (uint32_t)lutb[ab | ((b4hi >> (8 * j)) & 0xFFu)];
                }
            }
        }
        __syncthreads();            // protect As/Bs before next chunk's convert
    }

    // ---- epilogue: int -> float, add bias, store ----
    #pragma unroll
    for (int i = 0; i < 4; ++i) {
        const int m = m0 + mb + i;
        #pragma unroll
        for (int j = 0; j < 8; ++j) {
            const int n = n0 + nb + j;
            out[(size_t)m * N_DIM + n] = (float)acc[i][j] + bias[n];
        }
    }
}

extern "C" void kernel_launch(void* const* d_in, const int* in_sizes, int n_in,
                              void* d_out, int out_size, void* d_ws, size_t ws_size,
                              hipStream_t stream) {
    (void)in_sizes; (void)n_in; (void)d_ws; (void)ws_size; (void)out_size;
    const float* A    = (const float*)d_in[0];   // input  [M, K]
    const float* Wt   = (const float*)d_in[1];   // weight [N, K]
    const float* bias = (const float*)d_in[2];   // [N]
    const int*   lut  = (const int*)d_in[3];     // [256, 256] int32

    dim3 grid(N_DIM / BN, M_DIM / BM);           // (16, 16)
    dim3 block(NTHREADS);
    lut_dense_kernel<<<grid, block, SMEM_BYTES, stream>>>(A, Wt, bias, lut, (float*)d_out);
}